// _SSMCore_46643344834783
// MI455X (gfx1250) — compile-verified
//
#include <hip/hip_runtime.h>
#include <math.h>

// ---- problem constants (from reference) ----
#define D_INNER   2048
#define D_STATE   16
#define DT_RANK   64
#define BATCH     2
#define SEQ       2048
#define NROWS     (BATCH * SEQ)           // 4096 "M" rows (b*SEQ + t)
#define P_DIM     (DT_RANK + 2 * D_STATE) // 96
#define CHUNK     128                     // timesteps staged per TDM transfer
#define NCHUNK    (SEQ / CHUNK)           // 16

typedef __attribute__((ext_vector_type(2))) float v2f;
typedef __attribute__((ext_vector_type(8))) float v8f;
typedef __attribute__((ext_vector_type(4))) unsigned int v4u;
typedef __attribute__((ext_vector_type(4))) int  v4i;
typedef __attribute__((ext_vector_type(8))) int  v8i;

// =====================================================================
// Kernel 1: xz[M=4096][96] = x[M][2048] * x_proj_w[96][2048]^T
// One wave computes one 16x16 output tile with V_WMMA_F32_16X16X4_F32.
// =====================================================================
__global__ __launch_bounds__(256)
void ssm_proj1_wmma(const float* __restrict__ x,
                    const float* __restrict__ W,   // (96, 2048) row-major
                    float* __restrict__ xz)
{
    const int lane = threadIdx.x & 31;
    const int wave = threadIdx.x >> 5;
    const int tile = blockIdx.x * 8 + wave;       // 1536 tiles total
    const int mTiles = NROWS / 16;                // 256
    if (tile >= mTiles * (P_DIM / 16)) return;
    const int mt = tile % mTiles;
    const int nt = tile / mTiles;
    const int m0 = mt * 16;
    const int n0 = nt * 16;

    const int row  = lane & 15;
    const int half = lane >> 4;                   // 0: K 0..1, 1: K 2..3

    const float* aRow = x + (size_t)(m0 + row) * D_INNER + 2 * half;
    const float* bRow = W + (size_t)(n0 + row) * D_INNER + 2 * half;

    v8f c = {};
    #pragma unroll 8
    for (int k = 0; k < D_INNER; k += 4) {
        v2f a = *(const v2f*)(aRow + k);          // A[row][k+2h .. k+2h+1]
        v2f b = *(const v2f*)(bRow + k);          // B[k+2h..][n0+row] (col of W)
        c = __builtin_amdgcn_wmma_f32_16x16x4_f32(
                false, a, false, b, (short)0, c, false, false);
    }

    #pragma unroll
    for (int r = 0; r < 8; ++r) {
        xz[(size_t)(m0 + r + 8 * half) * P_DIM + n0 + row] = c[r];
    }
}

// =====================================================================
// Kernel 2: delta[M=4096][2048] = softplus( xz[:, :64] * dt_proj_w^T + b )
// =====================================================================
__global__ __launch_bounds__(256)
void ssm_proj2_wmma(const float* __restrict__ xz,
                    const float* __restrict__ Wd,   // (2048, 64)
                    const float* __restrict__ bias, // (2048,)
                    float* __restrict__ delta)
{
    const int lane = threadIdx.x & 31;
    const int wave = threadIdx.x >> 5;
    const int tile = blockIdx.x * 8 + wave;       // 256 * 128 = 32768 tiles
    const int mTiles = NROWS / 16;                // 256
    if (tile >= mTiles * (D_INNER / 16)) return;
    const int mt = tile % mTiles;
    const int nt = tile / mTiles;
    const int m0 = mt * 16;
    const int n0 = nt * 16;

    const int row  = lane & 15;
    const int half = lane >> 4;

    const float* aRow = xz + (size_t)(m0 + row) * P_DIM + 2 * half;
    const float* bRow = Wd + (size_t)(n0 + row) * DT_RANK + 2 * half;

    v8f c = {};
    #pragma unroll
    for (int k = 0; k < DT_RANK; k += 4) {
        v2f a = *(const v2f*)(aRow + k);
        v2f b = *(const v2f*)(bRow + k);
        c = __builtin_amdgcn_wmma_f32_16x16x4_f32(
                false, a, false, b, (short)0, c, false, false);
    }

    const float bn = bias[n0 + row];
    #pragma unroll
    for (int r = 0; r < 8; ++r) {
        float v = c[r] + bn;
        float sp = (v > 20.0f) ? v : log1pf(expf(v));   // softplus
        delta[(size_t)(m0 + r + 8 * half) * D_INNER + n0 + row] = sp;
    }
}

// =====================================================================
// TDM helper: async 2D tile load Global -> LDS (Tensor Data Mover).
// D# layout per CDNA5 ISA §8.3/8.4 (group0: count/lds/global/type,
// group1: data_size, tensor dims, tile dims, stride). 2-D tile, so
// groups 2/3 are zero. Tracked with TENSORcnt.
// =====================================================================
__device__ __forceinline__
void tdm_load_2d(unsigned int lds_addr, const float* gptr,
                 unsigned int width_elems, unsigned int rows,
                 unsigned int row_stride_elems)
{
    unsigned long long ga = (unsigned long long)(uintptr_t)gptr;
    v4u g0;
    g0.x = 1u;                                   // count=1, user descriptor
    g0.y = lds_addr;                             // LDS byte address
    g0.z = (unsigned int)ga;                     // global_addr[31:0]
    g0.w = (unsigned int)(ga >> 32) | (2u << 30); // global_addr[56:32] | type=2

    v8i g1;
    g1[0] = (int)(2u << 16);                              // data_size=2 (4B), mask=0
    g1[1] = (int)((width_elems & 0xFFFFu) << 16);         // tensor_dim0[15:0] @bit48
    g1[2] = (int)((width_elems >> 16) |
                  ((rows & 0xFFFFu) << 16));              // tdim0 hi | tensor_dim1 lo
    g1[3] = (int)((rows >> 16) |
                  ((width_elems & 0xFFFFu) << 16));       // tdim1 hi | tile_dim0
    g1[4] = (int)(rows & 0xFFFFu);                        // tile_dim1 | tile_dim2=0
    g1[5] = (int)row_stride_elems;                        // tensor_dim0_stride[31:0]
    g1[6] = 0;                                            // stride0 hi | stride1 lo
    g1[7] = 0;

    v4i z4 = {0, 0, 0, 0};
#if defined(__clang_major__) && (__clang_major__ >= 23)
    v8i z8 = {0, 0, 0, 0, 0, 0, 0, 0};
    __builtin_amdgcn_tensor_load_to_lds(g0, g1, z4, z4, z8, 0);
#else
    __builtin_amdgcn_tensor_load_to_lds(g0, g1, z4, z4, 0);
#endif
}

// =====================================================================
// Kernel 3: selective scan. One lane per (b, d, n) state element:
//   h = exp(delta*A[d,n]) * h + (delta*x) * B_t[n]
//   y[b,t,d] = sum_n h*C_t[n] + D[d]*x       (16-lane shfl_xor reduction)
// Per block: 16 channels of one batch. B_t/C_t (32 floats per t, shared
// block-wide) are staged in CHUNK-sized double-buffered LDS tiles via TDM.
// =====================================================================
__global__ __launch_bounds__(256)
void ssm_scan(const float* __restrict__ x,
              const float* __restrict__ xz,
              const float* __restrict__ delta,
              const float* __restrict__ A_log,
              const float* __restrict__ Dvec,
              float* __restrict__ out)
{
    __shared__ float bc[2][CHUNK * 2 * D_STATE];   // [slot][t*32 + {B:0..15,C:16..31}]

    const int tid = blockIdx.x * blockDim.x + threadIdx.x;  // 0 .. 65535
    const int n  = tid & (D_STATE - 1);
    const int ch = tid >> 4;                                 // 0 .. 4095
    const int d  = ch & (D_INNER - 1);
    const int b  = ch >> 11;                                 // D_INNER = 2^11

    const float A_n = -expf(A_log[(size_t)d * D_STATE + n]);
    const float Dd  = Dvec[d];

    const float* xz_bc = xz    + (size_t)b * SEQ * P_DIM + DT_RANK; // B cols @ t=0
    const float* x_b   = x     + (size_t)(b * SEQ) * D_INNER + d;
    const float* dlt_b = delta + (size_t)(b * SEQ) * D_INNER + d;
    float*       out_b = out   + (size_t)(b * SEQ) * D_INNER + d;

    const bool leader = (threadIdx.x < 32);   // wave 0 drives the TDM

    if (leader) {
        tdm_load_2d((unsigned int)(uintptr_t)&bc[0][0], xz_bc,
                    2 * D_STATE, CHUNK, P_DIM);
    }

    float h = 0.0f;
    for (int c = 0; c < NCHUNK; ++c) {
        if (leader) {
            __builtin_amdgcn_s_wait_tensorcnt(0);   // chunk c landed in LDS
        }
        __syncthreads();                            // publish LDS to all waves
        if (leader && (c + 1 < NCHUNK)) {
            tdm_load_2d((unsigned int)(uintptr_t)&bc[(c + 1) & 1][0],
                        xz_bc + (size_t)(c + 1) * CHUNK * P_DIM,
                        2 * D_STATE, CHUNK, P_DIM);
        }

        const float* bcc = &bc[c & 1][0];
        const int t0 = c * CHUNK;
        for (int tt = 0; tt < CHUNK; ++tt) {
            const int t = t0 + tt;
            if ((tt & 7) == 0) {   // stream-ahead hints -> global_prefetch_b8
                __builtin_prefetch(dlt_b + (size_t)(t + 16) * D_INNER, 0, 0);
                __builtin_prefetch(x_b   + (size_t)(t + 16) * D_INNER, 0, 0);
            }
            const float dlt = dlt_b[(size_t)t * D_INNER];
            const float xv  = x_b[(size_t)t * D_INNER];
            const float Bt  = bcc[tt * 2 * D_STATE + n];
            const float Ct  = bcc[tt * 2 * D_STATE + D_STATE + n];

            h = expf(dlt * A_n) * h + (dlt * xv) * Bt;

            float p = h * Ct;
            p += __shfl_xor(p, 8, 32);   // stays within each 16-lane half
            p += __shfl_xor(p, 4, 32);
            p += __shfl_xor(p, 2, 32);
            p += __shfl_xor(p, 1, 32);

            if (n == 0) out_b[(size_t)t * D_INNER] = p + Dd * xv;
        }
        __syncthreads();                // done reading slot before it is reused
    }
}

// =====================================================================
// Launcher
// inputs: 0:x 1:x_proj_w 2:dt_proj_w 3:dt_proj_b 4:A_log 5:D  (all fp32)
// output: y (BATCH, SEQ, D_INNER) fp32
// workspace: xz (4096*96 f32 = 1.5 MB) then delta (4096*2048 f32 = 32 MB)
// =====================================================================
extern "C" void kernel_launch(void* const* d_in, const int* in_sizes, int n_in,
                              void* d_out, int out_size, void* d_ws, size_t ws_size,
                              hipStream_t stream)
{
    const float* x     = (const float*)d_in[0];
    const float* xpw   = (const float*)d_in[1];
    const float* dtw   = (const float*)d_in[2];
    const float* dtb   = (const float*)d_in[3];
    const float* A_log = (const float*)d_in[4];
    const float* Dvec  = (const float*)d_in[5];
    float* out = (float*)d_out;

    float* xz    = (float*)d_ws;                         // 4096*96
    float* delta = xz + (size_t)NROWS * P_DIM;           // 4096*2048

    ssm_proj1_wmma<<<192, 256, 0, stream>>>(x, xpw, xz);
    ssm_proj2_wmma<<<4096, 256, 0, stream>>>(xz, dtw, dtb, delta);
    ssm_scan<<<256, 256, 0, stream>>>(x, xz, delta, A_log, Dvec, out);
}